// NequipBatch_73495480369506
// MI455X (gfx1250) — compile-verified
//
#include <hip/hip_runtime.h>
#include <math.h>

// Problem constants (match reference)
#define F_   8
#define N_   2048
#define E_   32768
#define C_   32
#define NT_  4
#define NB_  8
#define NSH_ 4
#define NL_  3
#define FN_  (F_ * N_)     // 16384 atoms
#define FE_  (F_ * E_)     // 262144 base edges (each processed in both directions)
#define R_MAX_ 6.0f

typedef float v2f __attribute__((ext_vector_type(2)));
typedef float v8f __attribute__((ext_vector_type(8)));

__device__ inline void atomAddF(float* p, float v) {
  // hardware global_atomic_add_f32
  unsafeAtomicAdd(p, v);
}

// ---------------------------------------------------------------------------
// Edge geometry: src/dst (frame-offset), PBC-shifted r and unit vector.
// Also zeroes the per-edge dE/dr accumulator.
// ---------------------------------------------------------------------------
__global__ void edge_setup(const float* __restrict__ pos,
                           const int* __restrict__ eidx,
                           const float* __restrict__ cell,
                           const int* __restrict__ cutp,
                           int* __restrict__ es, int* __restrict__ ed,
                           float* __restrict__ er, float* __restrict__ eu,
                           float* __restrict__ esacc) {
  int e = blockIdx.x * blockDim.x + threadIdx.x;
  if (e >= FE_) return;
  float co = (float)(*cutp);
  int f = e / E_, k = e - f * E_;
  int s = f * N_ + eidx[(f * 2 + 0) * E_ + k];
  int d = f * N_ + eidx[(f * 2 + 1) * E_ + k];
  float vx = pos[d * 3 + 0] - pos[s * 3 + 0];
  float vy = pos[d * 3 + 1] - pos[s * 3 + 1];
  float vz = pos[d * 3 + 2] - pos[s * 3 + 2];
  float cx = cell[f * 3 + 0], cy = cell[f * 3 + 1], cz = cell[f * 3 + 2];
  vx += cx * ((vx < -co ? 1.f : 0.f) - (vx > co ? 1.f : 0.f));
  vy += cy * ((vy < -co ? 1.f : 0.f) - (vy > co ? 1.f : 0.f));
  vz += cz * ((vz < -co ? 1.f : 0.f) - (vz > co ? 1.f : 0.f));
  float r = sqrtf(vx * vx + vy * vy + vz * vz + 1e-12f);
  float inv = 1.f / r;
  es[e] = s; ed[e] = d; er[e] = r;
  eu[e * 3 + 0] = vx * inv;
  eu[e * 3 + 1] = vy * inv;
  eu[e * 3 + 2] = vz * inv;
  esacc[e] = 0.f;
}

// ---------------------------------------------------------------------------
// Node init: feat0 = embedding gather, g = w_out broadcast (adjoint seed),
// zero d_out (energies + forces accumulated atomically).
// ---------------------------------------------------------------------------
__global__ void init_nodes(const int* __restrict__ at,
                           const float* __restrict__ Wemb,
                           const float* __restrict__ wout,
                           float* __restrict__ feat0,
                           float* __restrict__ g,
                           float* __restrict__ outp, int out_size) {
  int i = blockIdx.x * blockDim.x + threadIdx.x;
  if (i < FN_ * C_) {
    int n = i / C_, c = i - n * C_;
    feat0[i] = Wemb[at[n] * C_ + c];
    g[i] = wout[c];
  }
  if (i < out_size) outp[i] = 0.f;
}

__global__ void zero_buf(float* __restrict__ p, int n) {
  int i = blockIdx.x * blockDim.x + threadIdx.x;
  if (i < n) p[i] = 0.f;
}

// ---------------------------------------------------------------------------
// Bessel radial basis (scalar channel only) + analytic d/dr
// basis[b] = sqrt(2/Rmax) * env(r/Rmax) * sin((b+1)*pi*r/Rmax) / r
// ---------------------------------------------------------------------------
__device__ inline void compute_basis(float r, float* basis) {
  const float inv_rmax = 1.0f / R_MAX_;
  float x = r * inv_rmax;
  float env = 0.f;
  if (x < 1.f) {
    float x2 = x * x, x3 = x2 * x, x6 = x3 * x3;
    env = 1.f - 28.f * x6 + 48.f * x6 * x - 21.f * x6 * x2;
  }
  float pref = 0.5773502691896258f * env / r;   // sqrt(2/Rmax)=sqrt(1/3)
  float a = 3.14159265358979323846f * inv_rmax;
#pragma unroll
  for (int b = 0; b < NB_; b++)
    basis[b] = pref * __sinf((float)(b + 1) * a * r);
}

__device__ inline void compute_basis_deriv(float r, float* basis, float* dbasis) {
  const float inv_rmax = 1.0f / R_MAX_;
  float x = r * inv_rmax;
  float env = 0.f, denv = 0.f;
  if (x < 1.f) {
    float x2 = x * x, x3 = x2 * x, x5 = x3 * x2, x6 = x5 * x, x7 = x6 * x, x8 = x7 * x;
    env  = 1.f - 28.f * x6 + 48.f * x7 - 21.f * x8;
    denv = (-168.f * x5 + 336.f * x6 - 168.f * x7) * inv_rmax;  // d env / d r
  }
  const float Kc = 0.5773502691896258f;
  float invr = 1.f / r;
  float aa = 3.14159265358979323846f * inv_rmax;
#pragma unroll
  for (int b = 0; b < NB_; b++) {
    float a_b = (float)(b + 1) * aa;
    float ar = a_b * r;
    float S = __sinf(ar), Co = __cosf(ar);
    basis[b]  = Kc * env * S * invr;
    dbasis[b] = Kc * (denv * S * invr + env * (a_b * Co * r - S) * invr * invr);
  }
}

// ---------------------------------------------------------------------------
// Forward edge scatter, lane == channel mapping:
// one wave owns 8 edges; each atomic instruction's 32 lanes cover ONE
// contiguous 128B node row (single-cacheline atomic groups, coalesced gathers).
// W_rad scalar column lives in 8 regs/lane; edge meta broadcast via shfl.
// ---------------------------------------------------------------------------
__global__ void edge_fwd(const int* __restrict__ es, const int* __restrict__ ed,
                         const float* __restrict__ er,
                         const float* __restrict__ featL,
                         const float* __restrict__ WradL,
                         float* __restrict__ agg) {
  int lane = threadIdx.x & 31;
  int wave = (blockIdx.x * blockDim.x + threadIdx.x) >> 5;
  int eBase = wave * 8;
  float regW[NB_];
#pragma unroll
  for (int b = 0; b < NB_; b++)
    regW[b] = WradL[b * C_ * NSH_ + lane * NSH_];      // scalar (K=0) column
  int sl = 0, dl = 0; float rl = 0.f;
  if (lane < 8) {
    sl = es[eBase + lane];
    dl = ed[eBase + lane];
    rl = er[eBase + lane];
  }
#pragma unroll
  for (int i = 0; i < 8; i++) {
    int s   = __shfl(sl, i, 32);
    int d   = __shfl(dl, i, 32);
    float r = __shfl(rl, i, 32);
    float basis[NB_];
    compute_basis(r, basis);
    float R0 = 0.f;
#pragma unroll
    for (int b = 0; b < NB_; b++) R0 += basis[b] * regW[b];
    float fsv = featL[(size_t)s * C_ + lane];
    float fdv = featL[(size_t)d * C_ + lane];
    atomAddF(&agg[(size_t)d * C_ + lane], R0 * fsv);
    atomAddF(&agg[(size_t)s * C_ + lane], R0 * fdv);
  }
}

// ---------------------------------------------------------------------------
// Backward edge scatter (lane == channel) + per-edge dE/dr:
//   g[src] += R0*gA[dst], g[dst] += R0*gA[src]
//   dE/dr  = sum_c gR0[c] * dW[c],  dW[c] = sum_b dbasis[b]*W[b,c]
// (cross-lane sum via 5-step shfl_xor; lane 0 accumulates sacc[e]).
// ---------------------------------------------------------------------------
__global__ void edge_bwd(const int* __restrict__ es, const int* __restrict__ ed,
                         const float* __restrict__ er,
                         const float* __restrict__ featL,
                         const float* __restrict__ gA,
                         const float* __restrict__ WradL,
                         float* __restrict__ g,
                         float* __restrict__ sacc) {
  int lane = threadIdx.x & 31;
  int wave = (blockIdx.x * blockDim.x + threadIdx.x) >> 5;
  int eBase = wave * 8;
  float regW[NB_];
#pragma unroll
  for (int b = 0; b < NB_; b++)
    regW[b] = WradL[b * C_ * NSH_ + lane * NSH_];
  int sl = 0, dl = 0; float rl = 0.f;
  if (lane < 8) {
    sl = es[eBase + lane];
    dl = ed[eBase + lane];
    rl = er[eBase + lane];
  }
#pragma unroll
  for (int i = 0; i < 8; i++) {
    int s   = __shfl(sl, i, 32);
    int d   = __shfl(dl, i, 32);
    float r = __shfl(rl, i, 32);
    float basis[NB_], dbasis[NB_];
    compute_basis_deriv(r, basis, dbasis);
    float R0 = 0.f, dW = 0.f;
#pragma unroll
    for (int b = 0; b < NB_; b++) { R0 += basis[b] * regW[b]; dW += dbasis[b] * regW[b]; }
    float fsv  = featL[(size_t)s * C_ + lane];
    float fdv  = featL[(size_t)d * C_ + lane];
    float ga_s = gA[(size_t)s * C_ + lane];
    float ga_d = gA[(size_t)d * C_ + lane];
    atomAddF(&g[(size_t)s * C_ + lane], R0 * ga_d);
    atomAddF(&g[(size_t)d * C_ + lane], R0 * ga_s);
    float t = (ga_d * fsv + ga_s * fdv) * dW;          // per-channel dE/dr term
#pragma unroll
    for (int m = 16; m >= 1; m >>= 1) t += __shfl_xor(t, m, 32);
    if (lane == 0) sacc[eBase + i] += t;
  }
}

// ---------------------------------------------------------------------------
// Channel mix via V_WMMA_F32_16X16X4_F32 (templated: branch-free bodies):
//   Out = (RESID ? resid + : ) A @ B,  A: [FN_,32], B = W_mix (or W_mix^T).
// One wave per 16x16 output tile; all 16 fragment loads issued up front,
// then an 8-deep accumulator-chained WMMA sequence (K=32 as 8 x K=4).
// VGPR layouts per CDNA5 ISA 7.12.2. EXEC all-ones: exact tiling, no guards.
// ---------------------------------------------------------------------------
template <bool TRANSB, bool RESID>
__global__ void mix_wmma(const float* __restrict__ A,
                         const float* __restrict__ Wm,
                         const float* __restrict__ resid,
                         float* __restrict__ Out) {
  int wave = (blockIdx.x * blockDim.x + threadIdx.x) >> 5;
  int lane = threadIdx.x & 31;
  const int NTILES = C_ / 16;                 // 2
  int m0 = (wave / NTILES) * 16;
  int n0 = (wave % NTILES) * 16;
  int half = lane >> 4;                        // 0: K-low pair, 1: K-high pair
  int l15 = lane & 15;
  v2f a[8], b[8];
#pragma unroll
  for (int t = 0; t < 8; t++) {
    int ka = t * 4 + half * 2;                 // even -> 8B-aligned v2f loads
    a[t] = *(const v2f*)(A + (size_t)(m0 + l15) * C_ + ka);
    if (TRANSB) {                              // B[k][n] = Wm[n][k]
      b[t] = *(const v2f*)(Wm + (size_t)(n0 + l15) * C_ + ka);
    } else {                                   // B[k][n] = Wm[k][n]
      b[t].x = Wm[(size_t)(ka + 0) * C_ + n0 + l15];
      b[t].y = Wm[(size_t)(ka + 1) * C_ + n0 + l15];
    }
  }
  v8f acc = {};
#pragma unroll
  for (int t = 0; t < 8; t++) {
    acc = __builtin_amdgcn_wmma_f32_16x16x4_f32(
        /*neg_a=*/false, a[t], /*neg_b=*/false, b[t],
        /*c_mod=*/(short)0, acc, /*reuse_a=*/false, /*reuse_b=*/false);
  }
#pragma unroll
  for (int j = 0; j < 8; j++) {
    int row = m0 + j + half * 8;
    int col = n0 + l15;
    float v = acc[j];
    if (RESID) v += resid[(size_t)row * C_ + col];
    Out[(size_t)row * C_ + col] = v;
  }
}

// ---------------------------------------------------------------------------
// Per-frame energy: block-local reduce of feat3 . w_out, one atomic per block.
// Blocks never straddle frames (2048 % 256 == 0).
// ---------------------------------------------------------------------------
__global__ void energy_k(const float* __restrict__ feat3,
                         const float* __restrict__ wout,
                         float* __restrict__ out) {
  __shared__ float red[256];
  int n = blockIdx.x * 256 + threadIdx.x;
  float e = 0.f;
#pragma unroll
  for (int c = 0; c < C_; c++) e += feat3[(size_t)n * C_ + c] * wout[c];
  red[threadIdx.x] = e;
  __syncthreads();
  for (int s = 128; s > 0; s >>= 1) {
    if (threadIdx.x < s) red[threadIdx.x] += red[threadIdx.x + s];
    __syncthreads();
  }
  if (threadIdx.x == 0) atomAddF(&out[(blockIdx.x * 256) / N_], red[0]);
}

// ---------------------------------------------------------------------------
// Forces: F[src] += (dE/dr)*u, F[dst] -= (dE/dr)*u  (force = -grad)
// ---------------------------------------------------------------------------
__global__ void force_k(const int* __restrict__ es, const int* __restrict__ ed,
                        const float* __restrict__ eu,
                        const float* __restrict__ sacc,
                        float* __restrict__ out) {
  int e = blockIdx.x * blockDim.x + threadIdx.x;
  if (e >= FE_) return;
  float sv = sacc[e];
  float fx = sv * eu[e * 3 + 0];
  float fy = sv * eu[e * 3 + 1];
  float fz = sv * eu[e * 3 + 2];
  int s = es[e], d = ed[e];
  float* Fo = out + F_;                        // forces start after 8 energies
  atomAddF(&Fo[(size_t)s * 3 + 0],  fx);
  atomAddF(&Fo[(size_t)s * 3 + 1],  fy);
  atomAddF(&Fo[(size_t)s * 3 + 2],  fz);
  atomAddF(&Fo[(size_t)d * 3 + 0], -fx);
  atomAddF(&Fo[(size_t)d * 3 + 1], -fy);
  atomAddF(&Fo[(size_t)d * 3 + 2], -fz);
}

// ---------------------------------------------------------------------------
extern "C" void kernel_launch(void* const* d_in, const int* in_sizes, int n_in,
                              void* d_out, int out_size, void* d_ws, size_t ws_size,
                              hipStream_t stream) {
  const float* pos   = (const float*)d_in[0];
  const int*   eidx  = (const int*)  d_in[1];
  const float* cell  = (const float*)d_in[2];
  const int*   at    = (const int*)  d_in[3];
  const float* Wemb  = (const float*)d_in[4];
  const float* Wrad  = (const float*)d_in[5];
  const float* Wmix  = (const float*)d_in[6];
  const float* wout  = (const float*)d_in[7];
  const int*   cutp  = (const int*)  d_in[8];
  float* outp = (float*)d_out;

  // Workspace carve-out (~21 MB; fits easily in the 192 MB L2)
  size_t off = 0;
  char* ws = (char*)d_ws;
  auto carve = [&](size_t bytes) { void* p = ws + off; off += (bytes + 255) & ~(size_t)255; return p; };
  int*   es    = (int*)  carve((size_t)FE_ * 4);
  int*   ed    = (int*)  carve((size_t)FE_ * 4);
  float* er    = (float*)carve((size_t)FE_ * 4);
  float* eu    = (float*)carve((size_t)FE_ * 12);
  float* esacc = (float*)carve((size_t)FE_ * 4);
  float* feat  = (float*)carve((size_t)(NL_ + 1) * FN_ * C_ * 4);  // checkpoints 0..3
  float* agg   = (float*)carve((size_t)FN_ * C_ * 4);
  float* g     = (float*)carve((size_t)FN_ * C_ * 4);
  float* gA    = (float*)carve((size_t)FN_ * C_ * 4);
  (void)ws_size; (void)n_in; (void)in_sizes;

  const int nodeElems  = FN_ * C_;
  const int mixBlocks  = ((FN_ / 16) * (C_ / 16) * 32) / 256;  // 256 blocks
  const int edgeBlocks = FE_ / 64;                             // 8 edges/wave, 8 waves/blk

  edge_setup<<<FE_ / 256, 256, 0, stream>>>(pos, eidx, cell, cutp, es, ed, er, eu, esacc);
  init_nodes<<<nodeElems / 256, 256, 0, stream>>>(at, Wemb, wout, feat, g, outp, out_size);

  // Forward
  for (int l = 0; l < NL_; l++) {
    zero_buf<<<nodeElems / 256, 256, 0, stream>>>(agg, nodeElems);
    edge_fwd<<<edgeBlocks, 256, 0, stream>>>(es, ed, er,
        feat + (size_t)l * nodeElems, Wrad + (size_t)l * NB_ * C_ * NSH_, agg);
    mix_wmma<false, true><<<mixBlocks, 256, 0, stream>>>(agg,
        Wmix + (size_t)l * C_ * C_,
        feat + (size_t)l * nodeElems, feat + (size_t)(l + 1) * nodeElems);
  }

  energy_k<<<FN_ / 256, 256, 0, stream>>>(feat + (size_t)NL_ * nodeElems, wout, outp);

  // Backward (g seeded with w_out broadcast in init_nodes)
  for (int l = NL_ - 1; l >= 0; l--) {
    mix_wmma<true, false><<<mixBlocks, 256, 0, stream>>>(g,
        Wmix + (size_t)l * C_ * C_, nullptr, gA);
    edge_bwd<<<edgeBlocks, 256, 0, stream>>>(es, ed, er,
        feat + (size_t)l * nodeElems, gA, Wrad + (size_t)l * NB_ * C_ * NSH_, g, esacc);
  }

  force_k<<<FE_ / 256, 256, 0, stream>>>(es, ed, eu, esacc, outp);
}